// GraphModule_1571958030739
// MI455X (gfx1250) — compile-verified
//
#include <hip/hip_runtime.h>
#include <math.h>

#define N_NODES 20000
#define N_EDGES 320000
#define F_INCH  16
#define HIDC    64

typedef float v2f __attribute__((ext_vector_type(2)));
typedef float v8f __attribute__((ext_vector_type(8)));

// ---------------------------------------------------------------------------
// C[M,Nc] = A[M,K] * B[K,Nc]  (fp32, row-major, M%16==0, Nc%64==0, K%4==0)
// One wave per 16x64 strip: 4 V_WMMA_F32_16X16X4_F32 chains sharing one A frag.
// ---------------------------------------------------------------------------
__global__ void gemm_wmma_f32(const float* __restrict__ A,
                              const float* __restrict__ B,
                              float* __restrict__ C,
                              int M, int K, int Nc) {
    const int lane          = threadIdx.x & 31;
    const int wavesPerBlock = blockDim.x >> 5;
    const int wave          = blockIdx.x * wavesPerBlock + (threadIdx.x >> 5);
    const int tilesN4       = Nc >> 6;                  // 64-wide column strips
    const int totalTiles    = (M >> 4) * tilesN4;
    if (wave >= totalTiles) return;                     // wave-uniform branch

    const int tm   = wave / tilesN4;
    const int tn   = wave - tm * tilesN4;
    const int row  = (tm << 4) + (lane & 15);           // A row (M)
    const int colb = (tn << 6) + (lane & 15);           // first of 4 cols
    const int koff = (lane >> 4) << 1;                  // lanes 16-31: K+2,K+3

    v8f acc0 = {}, acc1 = {}, acc2 = {}, acc3 = {};
    const float* Arow = A + (size_t)row * K;
    for (int k = 0; k < K; k += 4) {
        v2f a;
        a.x = Arow[k + koff];
        a.y = Arow[k + koff + 1];
        const float* B0 = B + (size_t)(k + koff) * Nc;
        const float* B1 = B + (size_t)(k + koff + 1) * Nc;
        v2f b0, b1, b2, b3;
        b0.x = B0[colb];      b0.y = B1[colb];
        b1.x = B0[colb + 16]; b1.y = B1[colb + 16];
        b2.x = B0[colb + 32]; b2.y = B1[colb + 32];
        b3.x = B0[colb + 48]; b3.y = B1[colb + 48];
        acc0 = __builtin_amdgcn_wmma_f32_16x16x4_f32(false, a, false, b0, (short)0, acc0, false, false);
        acc1 = __builtin_amdgcn_wmma_f32_16x16x4_f32(false, a, false, b1, (short)0, acc1, false, false);
        acc2 = __builtin_amdgcn_wmma_f32_16x16x4_f32(false, a, false, b2, (short)0, acc2, false, false);
        acc3 = __builtin_amdgcn_wmma_f32_16x16x4_f32(false, a, false, b3, (short)0, acc3, false, false);
    }
    const int rbase = (tm << 4) + ((lane >> 4) << 3);   // C row for VGPR i
#pragma unroll
    for (int i = 0; i < 8; ++i) {
        float* Crow = C + (size_t)(rbase + i) * Nc;
        Crow[colb]      = acc0[i];
        Crow[colb + 16] = acc1[i];
        Crow[colb + 32] = acc2[i];
        Crow[colb + 48] = acc3[i];
    }
}

// ---------------------------------------------------------------------------
// Per-dst in-degree (int) + edge_attr sum (EDIM == 1)
// ---------------------------------------------------------------------------
__global__ void deg_sum_kernel(const int* __restrict__ dst,
                               const float* __restrict__ eattr,
                               int* __restrict__ cnt, float* __restrict__ suma,
                               int E) {
    int e = blockIdx.x * blockDim.x + threadIdx.x;
    if (e >= E) return;
    int d = dst[e];
    atomicAdd(&cnt[d], 1);
    atomicAdd(&suma[d], eattr[e]);
}

__global__ void mean_kernel(float* __restrict__ suma, const int* __restrict__ cnt, int n) {
    int i = blockIdx.x * blockDim.x + threadIdx.x;
    if (i >= n) return;
    suma[i] = suma[i] / fmaxf((float)cnt[i], 1.0f);
}

// ---------------------------------------------------------------------------
// Exclusive scan of (cnt[i] + 1) -> rowptr[0..n]  (single block, 1024 threads)
// (+1 accounts for each node's self-loop edge)
// ---------------------------------------------------------------------------
__global__ void scan_kernel(const int* __restrict__ cnt, int* __restrict__ rowptr, int n) {
    __shared__ int psum[1024];
    const int t = threadIdx.x;
    const int chunk = (n + 1023) / 1024;
    const int begin = t * chunk;
    const int end   = min(begin + chunk, n);
    int s = 0;
    for (int i = begin; i < end; ++i) s += cnt[i] + 1;
    psum[t] = s;
    __syncthreads();
    for (int o = 1; o < 1024; o <<= 1) {            // Hillis-Steele inclusive scan
        int v = (t >= o) ? psum[t - o] : 0;
        __syncthreads();
        psum[t] += v;
        __syncthreads();
    }
    int base = (t == 0) ? 0 : psum[t - 1];
    for (int i = begin; i < end; ++i) { rowptr[i] = base; base += cnt[i] + 1; }
    if (t == 1023) rowptr[n] = psum[1023];
}

// ---------------------------------------------------------------------------
// CSR fill: edge ids (real edges [0,E), virtual self loops [E,E+n)) per dst
// ---------------------------------------------------------------------------
__global__ void fill_kernel(const int* __restrict__ dstp, int* __restrict__ cursor,
                            const int* __restrict__ rowptr, int* __restrict__ eidx,
                            int E, int n) {
    int e = blockIdx.x * blockDim.x + threadIdx.x;
    if (e >= E + n) return;
    int d = (e < E) ? dstp[e] : e - E;
    int pos = atomicAdd(&cursor[d], 1);
    eidx[rowptr[d] + pos] = e;
}

// Sort each row ascending so accumulation order is deterministic.
__global__ void sort_rows_kernel(int* __restrict__ eidx, const int* __restrict__ rowptr, int n) {
    int d = blockIdx.x * blockDim.x + threadIdx.x;
    if (d >= n) return;
    int s = rowptr[d], e = rowptr[d + 1];
    for (int i = s + 1; i < e; ++i) {
        int v = eidx[i], j = i - 1;
        while (j >= s && eidx[j] > v) { eidx[j + 1] = eidx[j]; --j; }
        eidx[j + 1] = v;
    }
}

// ---------------------------------------------------------------------------
// Per-head scalar:  sh[h] = sum_c We[0, h*C+c] * ae[h,c]   (EDIM == 1)
// ---------------------------------------------------------------------------
__global__ void sh_kernel(const float* __restrict__ We, const float* __restrict__ ae,
                          float* __restrict__ sh, int H) {
    int h = threadIdx.x;
    if (h >= H) return;
    float s = 0.0f;
    for (int c = 0; c < HIDC; ++c) s += We[h * HIDC + c] * ae[h * HIDC + c];
    sh[h] = s;
}

// ---------------------------------------------------------------------------
// al_s[n,h] / al_d[n,h] = <h[n,h,:], a_src[h,:]> / <h[n,h,:], a_dst[h,:]>
// ---------------------------------------------------------------------------
__global__ void al_kernel(const float* __restrict__ hfeat,
                          const float* __restrict__ as_, const float* __restrict__ ad_,
                          float* __restrict__ al_s, float* __restrict__ al_d,
                          int n, int H) {
    int idx = blockIdx.x * blockDim.x + threadIdx.x;
    if (idx >= n * H) return;
    int node = idx / H, h = idx - node * H;
    const float* hv = hfeat + (size_t)node * H * HIDC + h * HIDC;
    float ss = 0.0f, sd = 0.0f;
    for (int c = 0; c < HIDC; ++c) {
        float v = hv[c];
        ss += v * as_[h * HIDC + c];
        sd += v * ad_[h * HIDC + c];
    }
    al_s[idx] = ss;
    al_d[idx] = sd;
}

// ---------------------------------------------------------------------------
// Fused GAT softmax + aggregation: one block per dst node, HC threads.
//   phase A: per-head segment max (LDS tree reduce)
//   phase B: per-head sum of exp
//   phase C: out[d,c] = sum_e h[src(e), c] * exp(alpha-m)/den  (+bias, opt ELU)
// No global atomics; deterministic (rows sorted).
// ---------------------------------------------------------------------------
template <int H, int HC>
__global__ void gat_node_kernel(const int* __restrict__ srcp, const int* __restrict__ eidx,
                                const int* __restrict__ rowptr,
                                const float* __restrict__ eattr, const float* __restrict__ meanat,
                                const float* __restrict__ al_s, const float* __restrict__ al_d,
                                const float* __restrict__ sh, const float* __restrict__ hfeat,
                                const float* __restrict__ bias, float* __restrict__ out,
                                int E, int applyElu) {
    const int d = blockIdx.x;
    const int t = threadIdx.x;
    const int start = rowptr[d];
    const int deg   = rowptr[d + 1] - start;

    __shared__ float red[H * HC];
    __shared__ float mh[H];
    __shared__ float dh[H];

    float shv[H], aldv[H];
#pragma unroll
    for (int h = 0; h < H; ++h) { shv[h] = sh[h]; aldv[h] = al_d[d * H + h]; }

    // ---- phase A: per-head max ----
    float lm[H];
#pragma unroll
    for (int h = 0; h < H; ++h) lm[h] = -INFINITY;
    for (int i = t; i < deg; i += HC) {
        int e = eidx[start + i];
        int s; float ea;
        if (e < E) { s = srcp[e]; ea = eattr[e]; } else { s = e - E; ea = meanat[s]; }
#pragma unroll
        for (int h = 0; h < H; ++h) {
            float a = al_s[s * H + h] + aldv[h] + ea * shv[h];
            a = a > 0.0f ? a : 0.2f * a;
            lm[h] = fmaxf(lm[h], a);
        }
    }
#pragma unroll
    for (int h = 0; h < H; ++h) red[h * HC + t] = lm[h];
    __syncthreads();
    for (int strd = HC >> 1; strd > 0; strd >>= 1) {
        if (t < strd)
#pragma unroll
            for (int h = 0; h < H; ++h)
                red[h * HC + t] = fmaxf(red[h * HC + t], red[h * HC + t + strd]);
        __syncthreads();
    }
    if (t < H) { float v = red[t * HC]; mh[t] = isfinite(v) ? v : 0.0f; }
    __syncthreads();

    // ---- phase B: per-head sum of exp ----
    float ls[H];
#pragma unroll
    for (int h = 0; h < H; ++h) ls[h] = 0.0f;
    for (int i = t; i < deg; i += HC) {
        int e = eidx[start + i];
        int s; float ea;
        if (e < E) { s = srcp[e]; ea = eattr[e]; } else { s = e - E; ea = meanat[s]; }
#pragma unroll
        for (int h = 0; h < H; ++h) {
            float a = al_s[s * H + h] + aldv[h] + ea * shv[h];
            a = a > 0.0f ? a : 0.2f * a;
            ls[h] += expf(a - mh[h]);
        }
    }
#pragma unroll
    for (int h = 0; h < H; ++h) red[h * HC + t] = ls[h];
    __syncthreads();
    for (int strd = HC >> 1; strd > 0; strd >>= 1) {
        if (t < strd)
#pragma unroll
            for (int h = 0; h < H; ++h)
                red[h * HC + t] += red[h * HC + t + strd];
        __syncthreads();
    }
    if (t < H) dh[t] = red[t * HC] + 1e-16f;
    __syncthreads();

    // ---- phase C: gather-aggregate (deterministic order) ----
    const int h = t / HIDC;
    const float invden = 1.0f / dh[h];
    const float mhv = mh[h], shh = shv[h], ald = aldv[h];
    float acc = 0.0f;
    for (int i = 0; i < deg; ++i) {
        int e = eidx[start + i];
        int s; float ea;
        if (e < E) { s = srcp[e]; ea = eattr[e]; } else { s = e - E; ea = meanat[s]; }
        float a = al_s[s * H + h] + ald + ea * shh;
        a = a > 0.0f ? a : 0.2f * a;
        float coef = expf(a - mhv) * invden;
        acc += hfeat[(size_t)s * HC + t] * coef;
    }
    float v = acc + bias[t];
    out[(size_t)d * HC + t] = applyElu ? (v > 0.0f ? v : (expf(v) - 1.0f)) : v;
}

// ---------------------------------------------------------------------------
// Final LayerNorm over 64 channels (bias already folded); one wave32 per node
// ---------------------------------------------------------------------------
__global__ void layernorm_kernel(const float* __restrict__ in,
                                 const float* __restrict__ g, const float* __restrict__ b,
                                 float* __restrict__ out, int n) {
    int node = blockIdx.x * (blockDim.x >> 5) + (threadIdx.x >> 5);
    int lane = threadIdx.x & 31;
    if (node >= n) return;
    float v0 = in[(size_t)node * HIDC + lane];
    float v1 = in[(size_t)node * HIDC + 32 + lane];
    float s = v0 + v1;
    for (int off = 16; off; off >>= 1) s += __shfl_xor(s, off, 32);
    float mu = s * (1.0f / 64.0f);
    float d0 = v0 - mu, d1 = v1 - mu;
    float vs = d0 * d0 + d1 * d1;
    for (int off = 16; off; off >>= 1) vs += __shfl_xor(vs, off, 32);
    float inv = rsqrtf(vs * (1.0f / 64.0f) + 1e-5f);
    out[(size_t)node * HIDC + lane]      = d0 * inv * g[lane]      + b[lane];
    out[(size_t)node * HIDC + 32 + lane] = d1 * inv * g[32 + lane] + b[32 + lane];
}

// ---------------------------------------------------------------------------
extern "C" void kernel_launch(void* const* d_in, const int* in_sizes, int n_in,
                              void* d_out, int out_size, void* d_ws, size_t ws_size,
                              hipStream_t stream) {
    (void)in_sizes; (void)n_in; (void)out_size; (void)ws_size;

    const float* x     = (const float*)d_in[0];
    const int*   ei    = (const int*)  d_in[1];
    const float* eattr = (const float*)d_in[2];
    const float* W0  = (const float*)d_in[3];
    const float* as0 = (const float*)d_in[4];
    const float* ad0 = (const float*)d_in[5];
    const float* We0 = (const float*)d_in[6];
    const float* ae0 = (const float*)d_in[7];
    const float* b0  = (const float*)d_in[8];
    const float* W1  = (const float*)d_in[9];
    const float* as1 = (const float*)d_in[10];
    const float* ad1 = (const float*)d_in[11];
    const float* We1 = (const float*)d_in[12];
    const float* ae1 = (const float*)d_in[13];
    const float* b1  = (const float*)d_in[14];
    const float* W2  = (const float*)d_in[15];
    const float* as2 = (const float*)d_in[16];
    const float* ad2 = (const float*)d_in[17];
    const float* We2 = (const float*)d_in[18];
    const float* ae2 = (const float*)d_in[19];
    const float* b2  = (const float*)d_in[20];
    const float* lng = (const float*)d_in[21];
    const float* lnb = (const float*)d_in[22];

    const int* srcp = ei;
    const int* dstp = ei + N_EDGES;
    const int  Et   = N_EDGES + N_NODES;

    // ---- workspace layout ----
    float* ws = (float*)d_ws;
    size_t off = 0;
    float* bufA   = ws + off; off += (size_t)N_NODES * 256;   // GEMM result
    float* bufB   = ws + off; off += (size_t)N_NODES * 256;   // agg L0 / L2
    float* bufC   = ws + off; off += (size_t)N_NODES * 256;   // agg L1
    float* al_s   = ws + off; off += (size_t)N_NODES * 4;
    float* al_d   = ws + off; off += (size_t)N_NODES * 4;
    float* meanat = ws + off; off += (size_t)N_NODES;
    float* shb    = ws + off; off += 8;
    int*   iw     = (int*)(ws + off);
    int*   cnt    = iw;                 iw += N_NODES;
    int*   cursor = iw;                 iw += N_NODES;
    int*   rowptr = iw;                 iw += N_NODES + 1;
    int*   eidx   = iw;                 iw += Et;

    // ---- one-time graph preprocessing (CSR over edges + self-loops) ----
    hipMemsetAsync(cnt,    0, (size_t)N_NODES * sizeof(int),   stream);
    hipMemsetAsync(cursor, 0, (size_t)N_NODES * sizeof(int),   stream);
    hipMemsetAsync(meanat, 0, (size_t)N_NODES * sizeof(float), stream);
    deg_sum_kernel<<<(N_EDGES + 255) / 256, 256, 0, stream>>>(dstp, eattr, cnt, meanat, N_EDGES);
    mean_kernel<<<(N_NODES + 255) / 256, 256, 0, stream>>>(meanat, cnt, N_NODES);
    scan_kernel<<<1, 1024, 0, stream>>>(cnt, rowptr, N_NODES);
    fill_kernel<<<(Et + 255) / 256, 256, 0, stream>>>(dstp, cursor, rowptr, eidx,
                                                      N_EDGES, N_NODES);
    sort_rows_kernel<<<(N_NODES + 255) / 256, 256, 0, stream>>>(eidx, rowptr, N_NODES);

    auto project = [&](const float* hin, int K, const float* W, float* hgemm, int HC) {
        const int totalTiles = (N_NODES / 16) * (HC / 64);
        const int wavesPerBlock = 8;                              // 256 threads
        gemm_wmma_f32<<<(totalTiles + wavesPerBlock - 1) / wavesPerBlock,
                        wavesPerBlock * 32, 0, stream>>>(hin, W, hgemm, N_NODES, K, HC);
    };

    // ---- layer 0: heads=4, in=16 ----
    project(x, F_INCH, W0, bufA, 256);
    sh_kernel<<<1, 32, 0, stream>>>(We0, ae0, shb, 4);
    al_kernel<<<(N_NODES * 4 + 255) / 256, 256, 0, stream>>>(bufA, as0, ad0, al_s, al_d, N_NODES, 4);
    gat_node_kernel<4, 256><<<N_NODES, 256, 0, stream>>>(srcp, eidx, rowptr, eattr, meanat,
                                                         al_s, al_d, shb, bufA, b0, bufB,
                                                         N_EDGES, 1);
    // ---- layer 1: heads=4, in=256 ----
    project(bufB, 256, W1, bufA, 256);
    sh_kernel<<<1, 32, 0, stream>>>(We1, ae1, shb, 4);
    al_kernel<<<(N_NODES * 4 + 255) / 256, 256, 0, stream>>>(bufA, as1, ad1, al_s, al_d, N_NODES, 4);
    gat_node_kernel<4, 256><<<N_NODES, 256, 0, stream>>>(srcp, eidx, rowptr, eattr, meanat,
                                                         al_s, al_d, shb, bufA, b1, bufC,
                                                         N_EDGES, 1);
    // ---- layer 2: heads=1, in=256 ----
    project(bufC, 256, W2, bufA, 64);
    sh_kernel<<<1, 32, 0, stream>>>(We2, ae2, shb, 1);
    al_kernel<<<(N_NODES + 255) / 256, 256, 0, stream>>>(bufA, as2, ad2, al_s, al_d, N_NODES, 1);
    gat_node_kernel<1, 64><<<N_NODES, 64, 0, stream>>>(srcp, eidx, rowptr, eattr, meanat,
                                                       al_s, al_d, shb, bufA, b2, bufB,
                                                       N_EDGES, 0);
    // ---- final layernorm -> d_out ----
    layernorm_kernel<<<(N_NODES + 7) / 8, 256, 0, stream>>>(bufB, lng, lnb,
                                                            (float*)d_out, N_NODES);
}